// NaCPG_41077067219466
// MI455X (gfx1250) — compile-verified
//
#include <hip/hip_runtime.h>
#include <cstdint>

// ---------------- problem constants ----------------
#define THREADS   256
#define EPT       16                 // edges per thread
#define CHUNK     (THREADS * EPT)    // 4096 edges per chunk (16 KB x2 per buffer)
#define PERSIST   625                // persistent blocks (3125 chunks = 625 x 5)

__constant__ const float kALPHA = 0.1f;
__constant__ const float kDT    = 0.01f;
__constant__ const float kEPS   = 1e-9f;
__constant__ const float kDIFF  = 10.0f;
__constant__ const float kLO    = -1.5707963267948966f;
__constant__ const float kHI    =  1.5707963267948966f;

typedef unsigned int v4u __attribute__((ext_vector_type(4)));
typedef int          v8i __attribute__((ext_vector_type(8)));
typedef int          v4i __attribute__((ext_vector_type(4)));

// ---------------------------------------------------------------------------
// TDM: 1-D tensor_load_to_lds of `nelem` dwords from `gaddr` into LDS byte
// offset `lds_off`.  D# per CDNA5 ISA ch.8.  6-arg builtin form
// (clang-23 / therock-10.0): (v4u, v8i, v4i, v4i, v8i, i32 cpol).
// ---------------------------------------------------------------------------
#if __has_builtin(__builtin_amdgcn_tensor_load_to_lds)
#define HAVE_TDM 1
__device__ __forceinline__ void tdm_load_dwords_to_lds(uint32_t lds_off,
                                                       const void* gaddr,
                                                       uint32_t nelem) {
    uint64_t ga = (uint64_t)(uintptr_t)gaddr;
    v4u g0;
    g0.x = 1u;                                    // count=1, user descriptor
    g0.y = lds_off;                               // lds_addr (bytes)
    g0.z = (uint32_t)(ga & 0xFFFFFFFFu);          // global_addr[31:0]
    g0.w = (uint32_t)((ga >> 32) & 0x01FFFFFFu)   // global_addr[56:32]
         | (2u << 30);                            // type = 2 ("image")
    v8i g1;
    g1[0] = (int)(2u << 16);                      // wg_mask=0, data_size=2 (4B)
    g1[1] = (int)((nelem & 0xFFFFu) << 16);       // tensor_dim0[15:0]
    g1[2] = (int)((nelem >> 16) | (1u << 16));    // tensor_dim0[31:16] | tensor_dim1=1
    g1[3] = (int)((nelem & 0xFFFFu) << 16);       // tile_dim0 (<=65535)
    g1[4] = 0;                                    // tile_dim1=0, tile_dim2=0
    g1[5] = (int)nelem;                           // tensor_dim0_stride[31:0]
    g1[6] = 0;
    g1[7] = 0;
    v4i gz4 = {0, 0, 0, 0};
    v8i gz8 = {0, 0, 0, 0, 0, 0, 0, 0};
    __builtin_amdgcn_tensor_load_to_lds(g0, g1, gz4, gz4, gz8, 0);
}
#else
#define HAVE_TDM 0
#endif

// ---------------------------------------------------------------------------
// Kernel A: per-node preprocessing.
//   PQ   = (cp*x + sp*y , sp*x - cp*y)     -> gathered per edge (L2-resident)
//   CPSP = (cos(phase), sin(phase))
//   K    = (a*x - b*y , b*x + a*y)
//   SPQ  = 0  (accumulators; re-zeroed every call for graph-replay safety)
// ---------------------------------------------------------------------------
__global__ __launch_bounds__(THREADS)
void node_prep(const float2* __restrict__ xy, const float2* __restrict__ xyd,
               const float*  __restrict__ phase, const float* __restrict__ w,
               const float*  __restrict__ ha,
               float2* __restrict__ PQ, float2* __restrict__ CPSP,
               float2* __restrict__ K,  float2* __restrict__ SPQ, int n) {
    int i = blockIdx.x * blockDim.x + threadIdx.x;
    if (i >= n) return;
    float2 p  = xy[i];
    float  x  = p.x, y = p.y;
    float  sp, cp;
    __sincosf(phase[i], &sp, &cp);
    PQ[i]   = make_float2(cp * x + sp * y, sp * x - cp * y);
    CPSP[i] = make_float2(cp, sp);
    float r2   = x * x + y * y;
    float a    = kALPHA * (1.0f - r2 * r2);
    float xdo  = xyd[i].x;
    float zeta = 1.0f - ha[i] * __fdividef(xdo + kEPS, fabsf(xdo) + kEPS);
    float b    = __fdividef(w[i], zeta + kEPS);
    K[i]   = make_float2(a * x - b * y, b * x + a * y);
    SPQ[i] = make_float2(0.0f, 0.0f);
}

// ---------------------------------------------------------------------------
// Per-chunk segmented gather-accumulate (sorted edge_src -> register runs,
// flush via f32 L2 atomics; ~1.25 flushes per 16 edges at avg degree 64).
// ---------------------------------------------------------------------------
__device__ __forceinline__ void process_chunk(const int* s_src, const int* s_dst,
                                              const float2* __restrict__ PQ,
                                              float* __restrict__ SPQ) {
    const int off = threadIdx.x * EPT;
    int ls[EPT], ld[EPT];
    const int4* s4 = (const int4*)&s_src[off];
    const int4* d4 = (const int4*)&s_dst[off];
#pragma unroll
    for (int j = 0; j < EPT / 4; ++j) {
        ((int4*)ls)[j] = s4[j];
        ((int4*)ld)[j] = d4[j];
    }
    int   cur = ls[0];
    float aP = 0.0f, aQ = 0.0f;
#pragma unroll
    for (int i = 0; i < EPT; ++i) {
        int    s  = ls[i];
        float2 pq = PQ[ld[i]];
        if (s != cur) {
            atomicAdd(&SPQ[2 * cur],     aP);
            atomicAdd(&SPQ[2 * cur + 1], aQ);
            cur = s; aP = 0.0f; aQ = 0.0f;
        }
        aP += pq.x;
        aQ += pq.y;
    }
    atomicAdd(&SPQ[2 * cur],     aP);
    atomicAdd(&SPQ[2 * cur + 1], aQ);
}

// ---------------------------------------------------------------------------
// Kernel B: persistent, double-buffered edge kernel.
// Wave 0 owns the TDM: it prefetches chunk c+gridDim into buf^1 while the
// block processes chunk c from buf.  Tensor ops complete in-order per wave,
// so after issuing the 2 new loads, s_wait_tensorcnt(2) waits exactly for
// the 2 older loads (current chunk).  TDM ignores EXEC, hence the whole-wave
// branch (threadIdx.x < 32) to keep it single-issue.
// ---------------------------------------------------------------------------
__global__ __launch_bounds__(THREADS)
void edge_kernel(const int* __restrict__ esrc, const int* __restrict__ edst,
                 const float2* __restrict__ PQ, float* __restrict__ SPQ,
                 int nchunks) {
    __shared__ __align__(16) int s_src[2][CHUNK];
    __shared__ __align__(16) int s_dst[2][CHUNK];

    const int c0 = blockIdx.x;
    if (c0 >= nchunks) return;

#if HAVE_TDM
    if (threadIdx.x < 32u) {   // wave 0: prefetch first chunk into buffer 0
        const long long b0 = (long long)c0 * CHUNK;
        tdm_load_dwords_to_lds((uint32_t)(uintptr_t)&s_src[0][0], esrc + b0, CHUNK);
        tdm_load_dwords_to_lds((uint32_t)(uintptr_t)&s_dst[0][0], edst + b0, CHUNK);
    }
    int buf = 0;
    for (int c = c0; c < nchunks; c += gridDim.x) {
        const int cn = c + (int)gridDim.x;
        if (threadIdx.x < 32u) {
            if (cn < nchunks) {
                const long long bn = (long long)cn * CHUNK;
                tdm_load_dwords_to_lds((uint32_t)(uintptr_t)&s_src[buf ^ 1][0],
                                       esrc + bn, CHUNK);
                tdm_load_dwords_to_lds((uint32_t)(uintptr_t)&s_dst[buf ^ 1][0],
                                       edst + bn, CHUNK);
                __builtin_amdgcn_s_wait_tensorcnt(2);  // older pair done
            } else {
                __builtin_amdgcn_s_wait_tensorcnt(0);
            }
        }
        __syncthreads();                       // data for chunk c visible
        process_chunk(s_src[buf], s_dst[buf], PQ, SPQ);
        __syncthreads();                       // all readers done before reuse
        buf ^= 1;
    }
#else
    for (int c = c0; c < nchunks; c += gridDim.x) {
        const long long base = (long long)c * CHUNK;
        for (int i = threadIdx.x; i < CHUNK; i += THREADS) {
            s_src[0][i] = esrc[base + i];
            s_dst[0][i] = edst[base + i];
        }
        __syncthreads();
        process_chunk(s_src[0], s_dst[0], PQ, SPQ);
        __syncthreads();
    }
#endif
}

// tail edges (only if n_edges % CHUNK != 0; exact for this problem)
__global__ __launch_bounds__(THREADS)
void edge_tail(const int* __restrict__ esrc, const int* __restrict__ edst,
               const float2* __restrict__ PQ, float* __restrict__ SPQ,
               long long start, long long n_edges) {
    long long e = start + (long long)blockIdx.x * blockDim.x + threadIdx.x;
    if (e >= n_edges) return;
    int s = esrc[e];
    float2 pq = PQ[edst[e]];
    atomicAdd(&SPQ[2 * s],     pq.x);
    atomicAdd(&SPQ[2 * s + 1], pq.y);
}

// ---------------------------------------------------------------------------
// Kernel C: rotate accumulated (ΣP,ΣQ) by source phase, clip, integrate.
//   sum_x = cp*ΣP + sp*ΣQ ; sum_y = sp*ΣP - cp*ΣQ
// ---------------------------------------------------------------------------
__global__ __launch_bounds__(THREADS)
void finalize(const float2* __restrict__ xy, const float2* __restrict__ xyd,
              const float* __restrict__ amp, const float2* __restrict__ CPSP,
              const float2* __restrict__ K,  const float2* __restrict__ SPQ,
              float* __restrict__ out_ang, float2* __restrict__ out_xy,
              float2* __restrict__ out_xydo, int n) {
    int i = blockIdx.x * blockDim.x + threadIdx.x;
    if (i >= n) return;
    float2 cs = CPSP[i];
    float2 S  = SPQ[i];
    float2 k  = K[i];
    float sum_x = cs.x * S.x + cs.y * S.y;
    float sum_y = cs.y * S.x - cs.x * S.y;
    float2 xo = xyd[i];
    float xdot = fminf(fmaxf(k.x + sum_x, xo.x - kDIFF), xo.x + kDIFF);
    float ydot = fminf(fmaxf(k.y + sum_y, xo.y - kDIFF), xo.y + kDIFF);
    float2 p  = xy[i];
    float xn = p.x + xdot * kDT;
    float yn = p.y + ydot * kDT;
    out_ang[i]  = fminf(fmaxf(amp[i] * yn, kLO), kHI);
    out_xy[i]   = make_float2(xn, yn);
    out_xydo[i] = p;   // xy_dot_old_new = original xy
}

// ---------------------------------------------------------------------------
extern "C" void kernel_launch(void* const* d_in, const int* in_sizes, int n_in,
                              void* d_out, int out_size, void* d_ws, size_t ws_size,
                              hipStream_t stream) {
    const float2* xy   = (const float2*)d_in[0];
    const float2* xyd  = (const float2*)d_in[1];
    const float*  ph   = (const float*) d_in[2];
    const float*  w    = (const float*) d_in[3];
    const float*  amp  = (const float*) d_in[4];
    const float*  ha   = (const float*) d_in[5];
    const int*    esrc = (const int*)   d_in[6];
    const int*    edst = (const int*)   d_in[7];

    const int       n       = in_sizes[2];            // N_NODES
    const long long n_edges = (long long)in_sizes[6]; // N_EDGES

    // workspace layout: PQ | CPSP | K | SPQ  (float2 each, n elements)
    float2* PQ   = (float2*)d_ws;
    float2* CPSP = PQ + n;
    float2* K    = CPSP + n;
    float2* SPQ  = K + n;

    float*  out      = (float*)d_out;
    float*  out_ang  = out;                  // n floats
    float2* out_xy   = (float2*)(out + n);   // n float2
    float2* out_xydo = (float2*)(out + 3 * (long long)n);

    const int nb = (n + THREADS - 1) / THREADS;
    node_prep<<<nb, THREADS, 0, stream>>>(xy, xyd, ph, w, ha, PQ, CPSP, K, SPQ, n);

    const long long full    = n_edges / CHUNK;        // 3125 for this problem
    const long long covered = full * CHUNK;
    if (full > 0) {
        int grid = (full > PERSIST) ? PERSIST : (int)full;
        edge_kernel<<<grid, THREADS, 0, stream>>>(esrc, edst, PQ, (float*)SPQ,
                                                  (int)full);
    }
    if (covered < n_edges) {
        long long rem = n_edges - covered;
        int tb = (int)((rem + THREADS - 1) / THREADS);
        edge_tail<<<tb, THREADS, 0, stream>>>(esrc, edst, PQ, (float*)SPQ,
                                              covered, n_edges);
    }

    finalize<<<nb, THREADS, 0, stream>>>(xy, xyd, amp, CPSP, K, SPQ,
                                         out_ang, out_xy, out_xydo, n);
}